// GraphAttentionLayer_75617194213830
// MI455X (gfx1250) — compile-verified
//
#include <hip/hip_runtime.h>
#include <hip/hip_bf16.h>
#include <math.h>

// Shapes: B=8, N=512, F=64, E=16, H=4, hd=16
// d_out layout (floats): out[262144] | e[33554432] | adj-bits[2097152]
// d_ws layout (floats):  h[262144] sr[16384] sc[16384] se[2097152] msg[262144]  (~10.6 MB)

typedef float v2f __attribute__((ext_vector_type(2)));
typedef float v8f __attribute__((ext_vector_type(8)));

__device__ __forceinline__ v8f wmma_f32_16x16x4(v2f a, v2f b, v8f c) {
#if defined(__gfx1250__)
  // D = A(16x4 f32) * B(4x16 f32) + C(16x16 f32), full f32 matrix path
  return __builtin_amdgcn_wmma_f32_16x16x4_f32(false, a, false, b, (short)0, c, false, false);
#else
  (void)a; (void)b; return c;
#endif
}

// branch-free CELU(alpha=1): x>0 ? x : exp(x)-1   (v_exp_f32 + cndmask, no EXEC branches)
__device__ __forceinline__ float celu1(float y) {
  float ey = __expf(y) - 1.0f;
  return (y > 0.0f) ? y : ey;
}

// ---------------------------------------------------------------------------
// Kernel 1: h = LN(node)@Wn^T + bn ; sr/sc head scores. One 64-thread block/row.
// ---------------------------------------------------------------------------
__global__ void k_node(const float* __restrict__ node,
                       const float* __restrict__ ln1_g, const float* __restrict__ ln1_b,
                       const float* __restrict__ Wn, const float* __restrict__ bn,
                       const float* __restrict__ attn_A,
                       float* __restrict__ h, float* __restrict__ sr, float* __restrict__ sc)
{
  const int row = blockIdx.x;   // b*512 + n
  const int t = threadIdx.x;    // 0..63
  __shared__ float red[64];
  __shared__ float xn[64];
  __shared__ float hr[64];

  float x = node[(size_t)row*64 + t];
  red[t] = x; __syncthreads();
  for (int s = 32; s > 0; s >>= 1) { if (t < s) red[t] += red[t+s]; __syncthreads(); }
  float mu = red[0] * (1.0f/64.0f); __syncthreads();
  float d = x - mu;
  red[t] = d*d; __syncthreads();
  for (int s = 32; s > 0; s >>= 1) { if (t < s) red[t] += red[t+s]; __syncthreads(); }
  float rs = rsqrtf(red[0]*(1.0f/64.0f) + 1e-5f); __syncthreads();
  xn[t] = d*rs*ln1_g[t] + ln1_b[t]; __syncthreads();

  float acc = bn[t];
  for (int k = 0; k < 64; k++) acc += xn[k]*Wn[t*64 + k];
  h[(size_t)row*64 + t] = acc; hr[t] = acc; __syncthreads();

  if (t < 4) {  // per-head row/col scores with a = attn_A.sum(0)
    float s1 = 0.0f, s2 = 0.0f;
    for (int x2 = 0; x2 < 16; x2++) {
      float ar = 0.0f, ac = 0.0f;
      for (int hh = 0; hh < 4; hh++) { ar += attn_A[hh*48 + x2]; ac += attn_A[hh*48 + 16 + x2]; }
      float hv = hr[t*16 + x2];
      s1 += hv*ar; s2 += hv*ac;
    }
    sr[(size_t)row*4 + t] = s1;
    sc[(size_t)row*4 + t] = s2;
  }
}

// ---------------------------------------------------------------------------
// Kernel 2 (bandwidth-dominant, fully fused, WMMA f32):
//   per 16-position tile of edge: LN(ln2) -> @We1^T+be1 -> celu -> @We2^T+be2
//   -> +edge residual -> LN(lne) -> store e, se = e . a_edge
// One wave32 per tile; 16 x V_WMMA_F32_16X16X4_F32 per tile.
// K-permutation trick: lane-half 0 owns K=0..7, half 1 owns K=8..15, so the
// GEMM1 A-fragments come straight from registers (no LDS staging, full-wave LN).
// ---------------------------------------------------------------------------
__global__ void __launch_bounds__(128) k_edge(
    const float* __restrict__ edge,
    const float* __restrict__ ln2_g, const float* __restrict__ ln2_b,
    const float* __restrict__ We1, const float* __restrict__ be1,
    const float* __restrict__ We2, const float* __restrict__ be2,
    const float* __restrict__ lne_g, const float* __restrict__ lne_b,
    const float* __restrict__ attn_A,
    float* __restrict__ e_out, float* __restrict__ se)
{
  const int lane = threadIdx.x & 31;
  const int wv   = threadIdx.x >> 5;
  const int tile = blockIdx.x*4 + wv;     // 131072 tiles total
  const int jt = tile & 31;               // 32 j-tiles per row
  const int i  = (tile >> 5) & 511;
  const int b  = tile >> 14;
  const int n    = lane & 15;             // row-within-tile / output column
  const int koff = lane >> 4;             // lane-half selector

  __shared__ float s_raw[4][16][17];
  __shared__ float s_buf[4][16][33];

  // ---- Preload B matrices (registers), WMMA-B 4x16 layout with permuted K ----
  // GEMM1 K-step kk covers physical K = { 2kk, 2kk+1 (half 0) ; 8+2kk, 8+2kk+1 (half 1) }
  // slot (koff, v) -> physical K = 8*koff + 2*kk + v
  v2f b1[2][4];
#pragma unroll
  for (int t = 0; t < 2; t++)
#pragma unroll
    for (int kk = 0; kk < 4; kk++) {
      int k = 8*koff + 2*kk;
      b1[t][kk].x = We1[(16*t + n)*16 + k];
      b1[t][kk].y = We1[(16*t + n)*16 + k + 1];
    }
  // GEMM2: standard K order; slot (koff, v) -> K = kk*4 + 2*koff + v
  v2f b2[8];
#pragma unroll
  for (int kk = 0; kk < 8; kk++) {
    int k = kk*4 + koff*2;
    b2[kk].x = We2[n*32 + k];
    b2[kk].y = We2[n*32 + k + 1];
  }

  const size_t tbase = ((((size_t)b*512 + i)*512) + (size_t)jt*16) * 16;
  const float* tp = edge + tbase;

  // ---- Stage A: LayerNorm(ln2); lane handles row n, K half [8*koff, 8*koff+8) ----
  float x[8];
  {
    const float4* rp4 = reinterpret_cast<const float4*>(tp + n*16 + koff*8);
    float4 v0 = rp4[0], v1 = rp4[1];
    x[0]=v0.x; x[1]=v0.y; x[2]=v0.z; x[3]=v0.w;
    x[4]=v1.x; x[5]=v1.y; x[6]=v1.z; x[7]=v1.w;
  }
  float ps = 0.0f;
#pragma unroll
  for (int j = 0; j < 8; j++) ps += x[j];
  float mu = (ps + __shfl_xor(ps, 16, 32)) * (1.0f/16.0f);
  float pv = 0.0f;
#pragma unroll
  for (int j = 0; j < 8; j++) { float d = x[j]-mu; pv += d*d; }
  float rs = rsqrtf((pv + __shfl_xor(pv, 16, 32)) * (1.0f/16.0f) + 1e-5f);
  float xn[8];
#pragma unroll
  for (int j = 0; j < 8; j++) {
    s_raw[wv][n][koff*8 + j] = x[j];
    xn[j] = (x[j]-mu)*rs*ln2_g[koff*8 + j] + ln2_b[koff*8 + j];
  }

  // ---- Stage B: GEMM1  Y1 = Xn(16x16) @ We1^T (16x32) + be1  (8 WMMAs) ----
  v8f c1[2];
#pragma unroll
  for (int t = 0; t < 2; t++) {
    float bias = be1[16*t + n];            // bias per output column (= lane&15)
#pragma unroll
    for (int v = 0; v < 8; v++) c1[t][v] = bias;
  }
#pragma unroll
  for (int kk = 0; kk < 4; kk++) {
    v2f a;                                  // from registers: local K idx 2kk, 2kk+1
    a.x = xn[2*kk];
    a.y = xn[2*kk + 1];
    c1[0] = wmma_f32_16x16x4(a, b1[0][kk], c1[0]);
    c1[1] = wmma_f32_16x16x4(a, b1[1][kk], c1[1]);
  }
  // CELU(alpha=1), branch-free
#pragma unroll
  for (int t = 0; t < 2; t++)
#pragma unroll
    for (int v = 0; v < 8; v++)
      c1[t][v] = celu1(c1[t][v]);

  // ---- Stage C: C-layout -> row-major LDS (transpose staging for GEMM2 A) ----
#pragma unroll
  for (int t = 0; t < 2; t++)
#pragma unroll
    for (int v = 0; v < 8; v++)
      s_buf[wv][v + 8*koff][16*t + n] = c1[t][v];   // element (M = v+8*half, N)
  __syncthreads();

  // ---- Stage D: GEMM2  Y2 = Y1(16x32) @ We2^T (32x16) + be2  (8 WMMAs) ----
  v8f c2;
  {
    float bias = be2[n];
#pragma unroll
    for (int v = 0; v < 8; v++) c2[v] = bias;
  }
#pragma unroll
  for (int kk = 0; kk < 8; kk++) {
    v2f a;
    a.x = s_buf[wv][n][kk*4 + koff*2];
    a.y = s_buf[wv][n][kk*4 + koff*2 + 1];
    c2 = wmma_f32_16x16x4(a, b2[kk], c2);
  }
  __syncthreads();

  // ---- Stage E: residual add, back to row-major ----
#pragma unroll
  for (int v = 0; v < 8; v++) {
    int M = v + 8*koff;
    s_buf[wv][M][n] = c2[v] + s_raw[wv][M][n];
  }
  __syncthreads();

  // ---- final LayerNorm(lne) + se dot + store (full wave: half-row per lane) ----
  {
    float y[8];
#pragma unroll
    for (int j = 0; j < 8; j++) y[j] = s_buf[wv][n][koff*8 + j];
    float ps2 = 0.0f;
#pragma unroll
    for (int j = 0; j < 8; j++) ps2 += y[j];
    float mu2 = (ps2 + __shfl_xor(ps2, 16, 32)) * (1.0f/16.0f);
    float pv2 = 0.0f;
#pragma unroll
    for (int j = 0; j < 8; j++) { float d = y[j]-mu2; pv2 += d*d; }
    float rs2 = rsqrtf((pv2 + __shfl_xor(pv2, 16, 32)) * (1.0f/16.0f) + 1e-5f);
    float pdot = 0.0f;
#pragma unroll
    for (int j = 0; j < 8; j++) {
      int k = koff*8 + j;
      float ae = attn_A[32+k] + attn_A[48+32+k] + attn_A[96+32+k] + attn_A[144+32+k];
      float z = (y[j]-mu2)*rs2*lne_g[k] + lne_b[k];
      y[j] = z;
      pdot += z*ae;
    }
    float4* op4 = reinterpret_cast<float4*>(e_out + tbase + (size_t)n*16 + koff*8);
    float4 o0, o1;
    o0.x=y[0]; o0.y=y[1]; o0.z=y[2]; o0.w=y[3];
    o1.x=y[4]; o1.y=y[5]; o1.z=y[6]; o1.w=y[7];
    op4[0] = o0; op4[1] = o1;
    float dot = pdot + __shfl_xor(pdot, 16, 32);
    if (koff == 0)
      se[(((size_t)b*512 + i)*512) + (size_t)jt*16 + n] = dot;
  }
}

// ---------------------------------------------------------------------------
// Kernel 3: masked leaky softmax over m (512) + msg aggregation.
// One wave32 per (b,n,head); butterfly shfl reductions (wave32 semantics).
// ---------------------------------------------------------------------------
__global__ void __launch_bounds__(256) k_attn(
    const float* __restrict__ h, const float* __restrict__ sr, const float* __restrict__ sc,
    const float* __restrict__ se, const int* __restrict__ adj,
    float* __restrict__ msg)
{
  const int lane = threadIdx.x & 31;
  const int wv   = threadIdx.x >> 5;
  const int item = blockIdx.x*8 + wv;   // 0 .. 16383
  const int hh   = item & 3;
  const int nrow = (item >> 2) & 511;
  const int b    = item >> 11;

  const float srv = sr[((size_t)b*512 + nrow)*4 + hh];
  const int*   adjrow = adj + (((size_t)b*512 + nrow)*512);
  const float* serow  = se  + (((size_t)b*512 + nrow)*512);

  float lg[16];
  float mx = -3.4e38f;
#pragma unroll
  for (int t = 0; t < 16; t++) {
    int m = lane + 32*t;
    float l = srv + sc[((size_t)b*512 + m)*4 + hh] + serow[m];
    if (adjrow[m] == 0) l = -1.0e9f;      // mask
    l = (l >= 0.0f) ? l : 0.2f*l;         // leaky_relu(0.2)
    l = fminf(1.0e9f, fmaxf(-1.0e9f, l)); // clip
    lg[t] = l;
    mx = fmaxf(mx, l);
  }
#pragma unroll
  for (int off = 16; off > 0; off >>= 1)
    mx = fmaxf(mx, __shfl_xor(mx, off, 32));

  float sum = 0.0f;
  float acc[16];
#pragma unroll
  for (int x = 0; x < 16; x++) acc[x] = 0.0f;
#pragma unroll
  for (int t = 0; t < 16; t++) {
    int m = lane + 32*t;
    float w = __expf(lg[t] - mx);         // arg <= 0, hardware v_exp_f32
    sum += w;
    const float4* hp = reinterpret_cast<const float4*>(h + ((size_t)b*512 + m)*64 + hh*16);
#pragma unroll
    for (int q = 0; q < 4; q++) {
      float4 v = hp[q];
      acc[q*4+0] += w*v.x; acc[q*4+1] += w*v.y;
      acc[q*4+2] += w*v.z; acc[q*4+3] += w*v.w;
    }
  }
#pragma unroll
  for (int off = 16; off > 0; off >>= 1) {
    sum += __shfl_xor(sum, off, 32);
#pragma unroll
    for (int x = 0; x < 16; x++) acc[x] += __shfl_xor(acc[x], off, 32);
  }
  if (lane == 0) {
    float inv = 1.0f/sum;   // sum>0 always (fully-masked row -> uniform, like ref)
    float4* mp4 = reinterpret_cast<float4*>(msg + ((size_t)b*512 + nrow)*64 + hh*16);
#pragma unroll
    for (int q = 0; q < 4; q++) {
      float4 v; v.x=acc[q*4]*inv; v.y=acc[q*4+1]*inv; v.z=acc[q*4+2]*inv; v.w=acc[q*4+3]*inv;
      mp4[q] = v;
    }
  }
}

// ---------------------------------------------------------------------------
// Kernel 4: out = LN( leaky( msg@Wo^T + bo + node ) ). One 64-thread block/row.
// ---------------------------------------------------------------------------
__global__ void k_out(const float* __restrict__ msg, const float* __restrict__ node,
                      const float* __restrict__ Wo, const float* __restrict__ bo,
                      const float* __restrict__ lno_g, const float* __restrict__ lno_b,
                      float* __restrict__ out)
{
  const int row = blockIdx.x;
  const int t = threadIdx.x;
  __shared__ float ms[64];
  __shared__ float red[64];

  ms[t] = msg[(size_t)row*64 + t];
  __syncthreads();
  float acc = bo[t];
  for (int k = 0; k < 64; k++) acc += ms[k]*Wo[t*64 + k];
  acc += node[(size_t)row*64 + t];
  acc = (acc >= 0.0f) ? acc : 0.2f*acc;

  red[t] = acc; __syncthreads();
  for (int s = 32; s > 0; s >>= 1) { if (t < s) red[t] += red[t+s]; __syncthreads(); }
  float mu = red[0]*(1.0f/64.0f); __syncthreads();
  float d = acc - mu;
  red[t] = d*d; __syncthreads();
  for (int s = 32; s > 0; s >>= 1) { if (t < s) red[t] += red[t+s]; __syncthreads(); }
  float rs = rsqrtf(red[0]*(1.0f/64.0f) + 1e-5f);
  out[(size_t)row*64 + t] = d*rs*lno_g[t] + lno_b[t];
}

// ---------------------------------------------------------------------------
extern "C" void kernel_launch(void* const* d_in, const int* in_sizes, int n_in,
                              void* d_out, int out_size, void* d_ws, size_t ws_size,
                              hipStream_t stream)
{
  const float* node  = (const float*)d_in[0];
  const float* edge  = (const float*)d_in[1];
  const int*   adj   = (const int*)  d_in[2];
  const float* ln1_g = (const float*)d_in[3];
  const float* ln1_b = (const float*)d_in[4];
  const float* Wn    = (const float*)d_in[5];
  const float* bn    = (const float*)d_in[6];
  const float* ln2_g = (const float*)d_in[7];
  const float* ln2_b = (const float*)d_in[8];
  const float* We1   = (const float*)d_in[9];
  const float* be1   = (const float*)d_in[10];
  const float* We2   = (const float*)d_in[11];
  const float* be2   = (const float*)d_in[12];
  const float* lne_g = (const float*)d_in[13];
  const float* lne_b = (const float*)d_in[14];
  const float* attn_A= (const float*)d_in[15];
  const float* Wo    = (const float*)d_in[16];
  const float* bo    = (const float*)d_in[17];
  const float* lno_g = (const float*)d_in[18];
  const float* lno_b = (const float*)d_in[19];

  float* out_node = (float*)d_out;             // 8*512*64
  float* out_e    = out_node + 262144;         // 8*512*512*16
  float* out_adj  = out_e + 33554432;          // adj bit-copied

  float* ws  = (float*)d_ws;
  float* h   = ws;                 // 262144
  float* sr  = h  + 262144;        // 16384
  float* sc  = sr + 16384;         // 16384
  float* se  = sc + 16384;         // 2097152
  float* msg = se + 2097152;       // 262144   (total ~10.6 MB)

  k_node<<<4096, 64, 0, stream>>>(node, ln1_g, ln1_b, Wn, bn, attn_A, h, sr, sc);
  k_edge<<<32768, 128, 0, stream>>>(edge, ln2_g, ln2_b, We1, be1, We2, be2,
                                    lne_g, lne_b, attn_A, out_e, se);
  k_attn<<<2048, 256, 0, stream>>>(h, sr, sc, se, adj, msg);
  k_out<<<4096, 64, 0, stream>>>(msg, node, Wo, bo, lno_g, lno_b, out_node);
  hipMemcpyAsync(out_adj, (const void*)adj, (size_t)2097152*sizeof(int),
                 hipMemcpyDeviceToDevice, stream);
}